// BSpline_16466904613521
// MI455X (gfx1250) — compile-verified
//
#include <hip/hip_runtime.h>

// B-spline basis expansion (degree 3, 64 basis fns, uniform knots on [-1,1]).
// Store-bound: 512MB out / 8MB in => ~22us floor at 23.3 TB/s HBM.
// Weights via V_WMMA_F32_16X16X4_F32 (16 rows/wave/WMMA); unconditional
// scatter into a padded per-wave LDS tile (D cols 4..15 are exact zeros, so
// they double as zero-fill); streamed out with NT b128 stores.

typedef __attribute__((ext_vector_type(2))) float v2f;
typedef __attribute__((ext_vector_type(4))) float v4f;
typedef __attribute__((ext_vector_type(8))) float v8f;

#define NBASIS 64
#define ROWS_PER_WAVE 16
#define WAVES_PER_BLOCK 8
#define ROWS_PER_BLOCK (ROWS_PER_WAVE * WAVES_PER_BLOCK)
#define PSTRIDE 84   // floats/row: 4 front guard + 64 valid + 16 tail guard (21 v4f)
#define PSTRIDE4 (PSTRIDE / 4)

__device__ __forceinline__ int interval_of(float xv) {
    xv = fminf(fmaxf(xv, -1.0f), 1.0f - 1e-7f);
    float u = (xv + 1.0f) * 33.5f;          // 1/h = 67/2
    int ii = (int)floorf(u);
    return max(0, min(66, ii));
}

__global__ __launch_bounds__(256)
void bspline_wmma_kernel(const float* __restrict__ x,
                         float* __restrict__ out,
                         int n)
{
    __shared__ float tile[WAVES_PER_BLOCK][ROWS_PER_WAVE * PSTRIDE]; // 42 KB

    const int lane = threadIdx.x & 31;
    const int nidx = lane & 15;   // D column / power-matrix row M
    const int half = lane >> 4;   // K-half (WMMA 32-bit A/B layout)
    const int wave = threadIdx.x >> 5;

    const int r0 = blockIdx.x * ROWS_PER_BLOCK + wave * ROWS_PER_WAVE;
    if (r0 >= n) return;          // wave-uniform: EXEC stays all-1s for WMMA

    const bool full = (r0 + ROWS_PER_WAVE <= n);   // wave-uniform

    // ---- scalar x for this lane's power-matrix row (row = nidx) ------------
    int rr = r0 + nidx; if (rr > n - 1) rr = n - 1;
    float xv = x[rr];
    xv = fminf(fmaxf(xv, -1.0f), 1.0f - 1e-7f);
    float u  = (xv + 1.0f) * 33.5f;
    int   ii = (int)floorf(u); ii = max(0, min(66, ii));
    float t  = u - (float)ii;

    // ---- interval indices for the 8 rows this lane scatters (rows 8*half+v)
    // 32 contiguous bytes of x (L0 hit); no cross-lane shuffles needed.
    int ii8[8];
    const int base = r0 + 8 * half;
    if (full) {
        v4f xa = *(const v4f*)(x + base);      // base % 8 == 0 -> 16B aligned
        v4f xb = *(const v4f*)(x + base + 4);
        ii8[0] = interval_of(xa.x); ii8[1] = interval_of(xa.y);
        ii8[2] = interval_of(xa.z); ii8[3] = interval_of(xa.w);
        ii8[4] = interval_of(xb.x); ii8[5] = interval_of(xb.y);
        ii8[6] = interval_of(xb.z); ii8[7] = interval_of(xb.w);
    } else {
#pragma unroll
        for (int v = 0; v < 8; ++v) {
            int r = base + v; if (r > n - 1) r = n - 1;
            ii8[v] = interval_of(x[r]);
        }
    }

    // ---- A: 16x4 powers of t (lanes 0-15: K=0,1 ; lanes 16-31: K=2,3) ------
    v2f a;
    if (half == 0) { a.x = 1.0f; a.y = t; }
    else           { float t2 = t * t; a.x = t2; a.y = t2 * t; }

    // ---- B: 4x16 cardinal-cubic coefficient bank in cols 0..3, zeros in 4..15
    //  w0=(1-t)^3/6  w1=(4-6t^2+3t^3)/6  w2=(1+3t+3t^2-3t^3)/6  w3=t^3/6
    float c0 = (nidx==0)?(1.0f/6.0f):(nidx==1)?(4.0f/6.0f):(nidx==2)?(1.0f/6.0f):0.0f;
    float c1 = (nidx==0)?-0.5f:(nidx==2)?0.5f:0.0f;
    float c2 = (nidx==0)?0.5f:(nidx==1)?-1.0f:(nidx==2)?0.5f:0.0f;
    float c3 = (nidx==0)?(-1.0f/6.0f):(nidx==1)?0.5f:(nidx==2)?-0.5f:(nidx==3)?(1.0f/6.0f):0.0f;
    v2f b;
    b.x = half ? c2 : c0;   // VGPR0: coeff rows K=0 (lanes 0-15) / K=2 (16-31)
    b.y = half ? c3 : c1;   // VGPR1: rows K=1 / K=3

    v8f cacc = {};
    v8f d = __builtin_amdgcn_wmma_f32_16x16x4_f32(
        /*neg_a=*/false, a, /*neg_b=*/false, b,
        /*c_mod=*/(short)0, cacc, /*reuse_a=*/false, /*reuse_b=*/false);

    // ---- zero the valid 64-col window of the padded per-wave tile ----------
    // idx = k*32+lane => row = 2k+half (stride 2*PSTRIDE4 v4f per k), c4 = nidx.
    float* wt  = tile[wave];
    v4f*   wt4 = (v4f*)wt;
    v4f*   zp  = wt4 + half * PSTRIDE4 + 1 + nidx;   // one base, const k-offsets
    v4f z = {0.f, 0.f, 0.f, 0.f};
#pragma unroll
    for (int k = 0; k < 8; ++k)
        zp[k * 2 * PSTRIDE4] = z;
    __builtin_amdgcn_wave_barrier();         // compiler fence; LDS in-order per wave

    // ---- unconditional scatter ---------------------------------------------
    // D lane mapping: lane L, reg v -> row v+8*(L>>4), col j=L&15.
    // Padded idx row*PSTRIDE + 4 + (i-3+j): j<4 = real weights at cols i-3..i;
    // j>=4 = exact WMMA zeros landing on must-be-zero cols i+1..i+12.
    // 16 distinct cols/row; guard padding absorbs out-of-window stores.
#pragma unroll
    for (int v = 0; v < 8; ++v) {
        int row = v + 8 * half;
        wt[row * PSTRIDE + 1 + ii8[v] + nidx] = d[v];   // 4 + (ii-3) + nidx
    }
    __builtin_amdgcn_wave_barrier();

    // ---- stream the 16x64 valid window out with NT b128 stores -------------
    v4f* src = zp;                                       // same LDS walk
    v4f* dst = (v4f*)(out + (size_t)r0 * NBASIS) + lane; // + k*32 per chunk
    if (full) {
#pragma unroll
        for (int k = 0; k < 8; ++k) {
            v4f vv = src[k * 2 * PSTRIDE4];
            __builtin_nontemporal_store(vv, dst + k * 32);
        }
    } else {
#pragma unroll
        for (int k = 0; k < 8; ++k) {
            int row = 2 * k + half;
            if (r0 + row < n) {
                v4f vv = src[k * 2 * PSTRIDE4];
                __builtin_nontemporal_store(vv, dst + k * 32);
            }
        }
    }
}

extern "C" void kernel_launch(void* const* d_in, const int* in_sizes, int n_in,
                              void* d_out, int out_size, void* d_ws, size_t ws_size,
                              hipStream_t stream) {
    const float* x = (const float*)d_in[0];
    float* out = (float*)d_out;
    int n = in_sizes[0];
    int blocks = (n + ROWS_PER_BLOCK - 1) / ROWS_PER_BLOCK;
    bspline_wmma_kernel<<<blocks, 256, 0, stream>>>(x, out, n);
}